// GINLayer_65240553226749
// MI455X (gfx1250) — compile-verified
//
#include <hip/hip_runtime.h>

#define NNODES 100000
#define NEDGES 1600000
#define DIM 128
#define NRB (NNODES / 16)   // 6250 row blocks of 16, exact

typedef __attribute__((ext_vector_type(2))) float v2f;
typedef __attribute__((ext_vector_type(8))) float v8f;

// ---------------------------------------------------------------------------
// Kernel 1: scatter-add. One wave (32 lanes) per edge; each lane moves 4
// floats (float4 coalesced read of feature[src], 4 f32 atomics into h[dst]).
// h was pre-initialized to feature, so after this kernel h = feature + neigh.
// ---------------------------------------------------------------------------
__global__ __launch_bounds__(256) void gin_scatter(
    const float* __restrict__ feature,
    const int* __restrict__ esrc,
    const int* __restrict__ edst,
    float* __restrict__ h) {
  long long gid = (long long)blockIdx.x * blockDim.x + threadIdx.x;
  int e = (int)(gid >> 5);            // wave-uniform
  if (e >= NEDGES) return;            // whole wave exits together
  int lane = (int)(gid & 31);
  int s = esrc[e];
  int d = edst[e];
  const float4 v = *(const float4*)(feature + (long long)s * DIM + lane * 4);
  float* p = h + (long long)d * DIM + lane * 4;
  atomicAdd(p + 0, v.x);
  atomicAdd(p + 1, v.y);
  atomicAdd(p + 2, v.z);
  atomicAdd(p + 3, v.w);
}

// ---------------------------------------------------------------------------
// Kernel 2: fused MLP out = relu(h@W1+b1)@W2+b2 using V_WMMA_F32_16X16X4_F32.
// One wave per 16-row block; 8 N-tiles (full 128-wide output) per wave.
// Layer-1 result is ReLU'd and staged row-major in a private 8KB LDS tile,
// then re-read as A-fragments for layer 2 (same wave -> no barrier needed).
// ---------------------------------------------------------------------------
__global__ __launch_bounds__(256) void gin_mlp(
    const float* __restrict__ h,
    const float* __restrict__ W1, const float* __restrict__ b1,
    const float* __restrict__ W2, const float* __restrict__ b2,
    float* __restrict__ out) {
  __shared__ float smem[8 * 16 * DIM];  // 64 KB: 8 waves x (16x128) staging

  const int lane = threadIdx.x & 31;
  const int wave = threadIdx.x >> 5;
  const int rb = blockIdx.x * 8 + wave;
  if (rb >= NRB) return;               // wave-uniform; EXEC all-1s below

  const int lrow  = lane & 15;         // N (for B/C/D) or M (for A)
  const int khalf = (lane >> 4) << 1;  // K sub-offset: 0 or 2
  const int mhalf = (lane >> 4) << 3;  // M sub-offset for C/D: 0 or 8
  float* mid = smem + wave * (16 * DIM);

  // ---- Layer 1: acc = h_tile @ W1 + b1 ----
  v8f acc[8];
#pragma unroll
  for (int nt = 0; nt < 8; ++nt) {
    float bv = b1[nt * 16 + lrow];
    acc[nt] = (v8f){bv, bv, bv, bv, bv, bv, bv, bv};
  }
  const float* hrow = h + ((long long)rb * 16 + lrow) * DIM;
  for (int k0 = 0; k0 < DIM; k0 += 4) {
    // A frag: lane l holds h[row0 + l%16][k0 + (l/16)*2 + j], j=0,1
    v2f a = *(const v2f*)(hrow + k0 + khalf);
#pragma unroll
    for (int nt = 0; nt < 8; ++nt) {
      const int col = nt * 16 + lrow;
      v2f b;
      b.x = W1[(k0 + khalf) * DIM + col];
      b.y = W1[(k0 + khalf + 1) * DIM + col];
      acc[nt] = __builtin_amdgcn_wmma_f32_16x16x4_f32(
          false, a, false, b, (short)0, acc[nt], false, false);
    }
  }

  // ---- ReLU + stage layer-1 tile to LDS in row-major order ----
#pragma unroll
  for (int nt = 0; nt < 8; ++nt) {
#pragma unroll
    for (int r = 0; r < 8; ++r) {
      float v = acc[nt][r];
      mid[(r + mhalf) * DIM + nt * 16 + lrow] = v > 0.0f ? v : 0.0f;
    }
  }

  // ---- Layer 2: acc2 = mid @ W2 + b2 ----
  v8f acc2[8];
#pragma unroll
  for (int nt = 0; nt < 8; ++nt) {
    float bv = b2[nt * 16 + lrow];
    acc2[nt] = (v8f){bv, bv, bv, bv, bv, bv, bv, bv};
  }
  for (int k0 = 0; k0 < DIM; k0 += 4) {
    v2f a = *(const v2f*)(mid + lrow * DIM + k0 + khalf);  // ds_load_b64
#pragma unroll
    for (int nt = 0; nt < 8; ++nt) {
      const int col = nt * 16 + lrow;
      v2f b;
      b.x = W2[(k0 + khalf) * DIM + col];
      b.y = W2[(k0 + khalf + 1) * DIM + col];
      acc2[nt] = __builtin_amdgcn_wmma_f32_16x16x4_f32(
          false, a, false, b, (short)0, acc2[nt], false, false);
    }
  }

  // ---- Store 16x128 output tile ----
  float* orow = out + (long long)rb * 16 * DIM;
#pragma unroll
  for (int nt = 0; nt < 8; ++nt) {
#pragma unroll
    for (int r = 0; r < 8; ++r) {
      orow[(r + mhalf) * DIM + nt * 16 + lrow] = acc2[nt][r];
    }
  }
}

extern "C" void kernel_launch(void* const* d_in, const int* in_sizes, int n_in,
                              void* d_out, int out_size, void* d_ws, size_t ws_size,
                              hipStream_t stream) {
  const float* feature = (const float*)d_in[0];
  const int*   esrc    = (const int*)d_in[1];
  const int*   edst    = (const int*)d_in[2];
  const float* W1      = (const float*)d_in[3];
  const float* b1      = (const float*)d_in[4];
  const float* W2      = (const float*)d_in[5];
  const float* b2      = (const float*)d_in[6];
  float* out  = (float*)d_out;
  float* hbuf = (float*)d_ws;  // N x D fp32 = 51.2 MB

  // h <- feature (fuses the "(1+eps)*feature" combine; eps == 0)
  hipMemcpyAsync(hbuf, feature, (size_t)NNODES * DIM * sizeof(float),
                 hipMemcpyDeviceToDevice, stream);

  // h += segment_sum(feature[src] -> dst)
  {
    long long threads = (long long)NEDGES * 32;
    int block = 256;
    unsigned grid = (unsigned)((threads + block - 1) / block);  // 200000
    gin_scatter<<<grid, block, 0, stream>>>(feature, esrc, edst, hbuf);
  }

  // out = relu(h@W1+b1)@W2+b2
  {
    int blocks = (NRB + 7) / 8;  // 782 blocks x 8 waves, tail wave-uniform
    gin_mlp<<<blocks, 256, 0, stream>>>(hbuf, W1, b1, W2, b2, out);
  }
}